// GraphNoiseModel_39754217292305
// MI455X (gfx1250) — compile-verified
//
#include <hip/hip_runtime.h>
#include <math.h>

// Collapsed GraphNoiseModel:
//   out[b,j] = exp(log_amp[j,month]) *
//              softplus( dot( relu(x_b @ W1[0:128] + emb*W1[128] + b1), rowsum(W2) )/128 + mean(b2) )
// The GCN aggregation is the identity because every destination's incoming
// norm-sum is exactly 1 and all nodes carry the same feature row.

typedef __attribute__((ext_vector_type(2))) float v2f;
typedef __attribute__((ext_vector_type(8))) float v8f;

#define SD      128   // STATE_DIM
#define BATCH   512
#define HID     32
#define THREADS 256
#define WAVES   8           // 256 / wave32
#define ROWS_PB (WAVES * 16)  // 128 batch rows per block

__launch_bounds__(THREADS)
__global__ void gnm_fused_kernel(const float* __restrict__ x,       // [512,128]
                                 const float* __restrict__ t_,      // [1]
                                 const float* __restrict__ log_amp, // [128,12]
                                 const float* __restrict__ W1,      // [129,32]
                                 const float* __restrict__ b1,      // [32]
                                 const float* __restrict__ W2,      // [32,128]
                                 const float* __restrict__ b2,      // [128]
                                 float* __restrict__ out)           // [512,128]
{
    __shared__ float s_y[WAVES][16][HID];   // per-wave 16x32 y tiles (16 KB)
    __shared__ float s_amp[SD];             // seasonal amplitude
    __shared__ float s_hbias[HID];          // emb*W1[128,:] + b1
    __shared__ float s_w2sum[HID];          // rowsum(W2)
    __shared__ float s_s[ROWS_PB];          // softplus scalar per batch row
    __shared__ float s_meanb2;

    const int tid  = threadIdx.x;
    const int wave = tid >> 5;          // wave32
    const int lane = tid & 31;
    const int rowbase = blockIdx.x * ROWS_PB;

    // ---------------- preamble: tiny per-block scalars ----------------
    const float t = t_[0];
    if (tid < SD) {
        int month = (int)fmodf(t, 12.0f);           // t in [0,24)
        s_amp[tid] = expf(log_amp[tid * 12 + month]);
    } else if (tid < SD + HID) {
        int n = tid - SD;
        float emb = sinf(2.0f * 3.14159265358979323846f * t * (1.0f / 12.0f));
        s_hbias[n] = emb * W1[SD * HID + n] + b1[n];  // W1 row 128 folded in
    } else if (tid < SD + 2 * HID) {
        int n = tid - SD - HID;
        float acc = 0.0f;
        #pragma unroll 8
        for (int f = 0; f < SD; ++f) acc += W2[n * SD + f];
        s_w2sum[n] = acc;
    } else if (tid == SD + 2 * HID) {
        float acc = 0.0f;
        #pragma unroll 8
        for (int f = 0; f < SD; ++f) acc += b2[f];
        s_meanb2 = acc * (1.0f / (float)SD);
    }
    // (no barrier needed yet: preamble results consumed only after the
    //  post-GEMM barrier below; EXEC reconverges here -> WMMA legal)

    // ---------------- WMMA GEMM: y[16x32] = x_tile[16x128] @ W1[128x32] ----
    // A (16x4 f32): row = lane&15, VGPR0/1 = K = (lane>>4)*2, +1
    // B (4x16 f32): col = lane&15, lanes 0-15 -> K rows 0..1, lanes 16-31 -> K rows 2..3
    const int mrow  = lane & 15;
    const int khalf = (lane >> 4) * 2;
    const int ncol  = lane & 15;
    const float* aptr = x + (size_t)(rowbase + wave * 16 + mrow) * SD + khalf;

    v8f c0 = {};   // N = 0..15
    v8f c1 = {};   // N = 16..31

    #pragma unroll 4
    for (int k = 0; k < SD; k += 4) {
        v2f a = *(const v2f*)(aptr + k);                 // 8B-aligned (k+khalf even)
        const float* bp = W1 + (size_t)(k + khalf) * HID;
        v2f b0; b0.x = bp[ncol];      b0.y = bp[HID + ncol];
        v2f b1v; b1v.x = bp[ncol + 16]; b1v.y = bp[HID + ncol + 16];
        c0 = __builtin_amdgcn_wmma_f32_16x16x4_f32(false, a, false, b0,
                                                   (short)0, c0, false, false);
        c1 = __builtin_amdgcn_wmma_f32_16x16x4_f32(false, a, false, b1v,
                                                   (short)0, c1, false, false);
    }

    // C/D layout: VGPR r holds M = (lane>>4)*8 + r, N = lane&15
    {
        const int mbase = (lane >> 4) * 8;
        #pragma unroll
        for (int r = 0; r < 8; ++r) {
            s_y[wave][mbase + r][ncol]      = c0[r];
            s_y[wave][mbase + r][ncol + 16] = c1[r];
        }
    }
    __syncthreads();

    // ---------------- epilogue 1: per-row scalar ----------------
    if (tid < ROWS_PB) {
        const int w = tid >> 4;
        const int r = tid & 15;
        float acc = 0.0f;
        #pragma unroll
        for (int n = 0; n < HID; ++n) {
            float h = s_y[w][r][n] + s_hbias[n];
            h = fmaxf(h, 0.0f);                 // relu
            acc += h * s_w2sum[n];
        }
        float m  = acc * (1.0f / (float)SD) + s_meanb2;
        // numerically stable softplus
        s_s[tid] = fmaxf(m, 0.0f) + log1pf(expf(-fabsf(m)));
    }
    __syncthreads();

    // ---------------- epilogue 2: out[b,:] = amp[:] * s_b (float4 stores) ---
    for (int idx = tid; idx < ROWS_PB * (SD / 4); idx += THREADS) {
        const int row = idx >> 5;           // SD/4 == 32 vec4 per row
        const int c4  = (idx & 31) * 4;
        const float sp = s_s[row];
        float4 v;
        v.x = sp * s_amp[c4 + 0];
        v.y = sp * s_amp[c4 + 1];
        v.z = sp * s_amp[c4 + 2];
        v.w = sp * s_amp[c4 + 3];
        *(float4*)(out + (size_t)(rowbase + row) * SD + c4) = v;
    }
}

extern "C" void kernel_launch(void* const* d_in, const int* in_sizes, int n_in,
                              void* d_out, int out_size, void* d_ws, size_t ws_size,
                              hipStream_t stream) {
    (void)in_sizes; (void)n_in; (void)out_size; (void)d_ws; (void)ws_size;
    const float* x       = (const float*)d_in[0];
    const float* t       = (const float*)d_in[1];
    const float* log_amp = (const float*)d_in[2];
    const float* W1      = (const float*)d_in[3];
    const float* b1      = (const float*)d_in[4];
    const float* W2      = (const float*)d_in[5];
    const float* b2      = (const float*)d_in[6];
    float* out           = (float*)d_out;

    dim3 grid(BATCH / ROWS_PB);   // 4 blocks
    dim3 block(THREADS);          // 8 wave32 waves
    gnm_fused_kernel<<<grid, block, 0, stream>>>(x, t, log_amp, W1, b1, W2, b2, out);
}